// Aleat5_1ChamferL2NormalWeightedALLSVDLoss_30176440222102
// MI455X (gfx1250) — compile-verified
//
#include <hip/hip_runtime.h>
#include <math.h>
#include <float.h>

#define NBATCH 2
#define REG_W 0.001f
#define LOSS_W 1.0f
#define EPS_COS 1e-8f

typedef __attribute__((ext_vector_type(2))) float v2f;
typedef __attribute__((ext_vector_type(8))) float v8f;

// ---------------------------------------------------------------------------
// Kernel 0: zero the per-batch accumulators
// ---------------------------------------------------------------------------
__global__ void init_acc_kernel(float* acc, int n) {
    int i = blockIdx.x * blockDim.x + threadIdx.x;
    if (i < n) acc[i] = 0.0f;
}

// ---------------------------------------------------------------------------
// Kernel 1: Chamfer NN search via V_WMMA_F32_16X16X4_F32.
// One wave owns a 16-row tile of xp; xt (+ its squared norms) is staged in LDS
// once per block. The WMMA computes the FULL distance:
//   A = (-2x, -2y, -2z, 1)  (16x4, f32)
//   B = ( x',  y',  z', |xt'|^2 )  (4x16, f32)
//   C = |xp|^2 broadcast per row
//   D = |xp|^2 - 2 xp.xt' + |xt'|^2  == d2   (no post-WMMA arithmetic)
// so the per-tile VALU work is only the (min, argmin) selection.
// ---------------------------------------------------------------------------
#define CH_WAVES 8
__global__ void chamfer_wmma_kernel(const float* __restrict__ xyz_pred,
                                    const float* __restrict__ xyz_target,
                                    float* __restrict__ diff,
                                    int* __restrict__ idxb,
                                    int N, int M) {
    extern __shared__ float smem[];          // [3*M] xt  +  [M] |xt|^2
    float* sxt = smem;
    float* snt = smem + 3 * M;

    const int b = blockIdx.y;
    const float* xp = xyz_pred   + (size_t)b * N * 3;
    const float* xt = xyz_target + (size_t)b * M * 3;

    // --- stage xt into LDS (vectorized) ---
    {
        const int totalF4 = (3 * M) / 4;     // 3*M multiple of 4
        const float4* g4 = (const float4*)xt;
        float4* s4 = (float4*)sxt;
        for (int i = threadIdx.x; i < totalF4; i += blockDim.x) s4[i] = g4[i];
    }
    __syncthreads();
    for (int j = threadIdx.x; j < M; j += blockDim.x) {
        float tx = sxt[3 * j], ty = sxt[3 * j + 1], tz = sxt[3 * j + 2];
        snt[j] = tx * tx + ty * ty + tz * tz;
    }
    __syncthreads();

    const int wave = threadIdx.x >> 5;
    const int lane = threadIdx.x & 31;
    const int hi   = lane >> 4;              // 0: K=0,1 half  1: K=2,3 half
    const int l    = lane & 15;
    const int i0   = (blockIdx.x * CH_WAVES + wave) * 16;

    // A fragment: 16x4 f32, coords pre-scaled by -2, K=3 column = 1
    v2f afrag;
    {
        const float* p = xp + (size_t)(i0 + l) * 3;
        afrag.x = hi ? (-2.0f * p[2]) : (-2.0f * p[0]);
        afrag.y = hi ? 1.0f           : (-2.0f * p[1]);
    }

    // C fragment: |xp|^2 for the 8 rows this lane's registers cover
    v8f cinit;
#pragma unroll
    for (int v = 0; v < 8; ++v) {
        const float* p = xp + (size_t)(i0 + v + 8 * hi) * 3;
        cinit[v] = p[0] * p[0] + p[1] * p[1] + p[2] * p[2];
    }

    float minv[8];
    int   mini[8];
#pragma unroll
    for (int v = 0; v < 8; ++v) { minv[v] = FLT_MAX; mini[v] = 0; }

#pragma unroll 4
    for (int j0 = 0; j0 < M; j0 += 16) {
        const int col = j0 + l;
        const float tx = sxt[3 * col];
        const float ty = sxt[3 * col + 1];
        const float tz = sxt[3 * col + 2];
        const float nt = snt[col];
        v2f bfrag;
        bfrag.x = hi ? tz : tx;              // K=0 / K=2 rows of B
        bfrag.y = hi ? nt : ty;              // K=1 / K=3(|xt|^2) rows of B

        v8f c = __builtin_amdgcn_wmma_f32_16x16x4_f32(
                false, afrag, false, bfrag, (short)0, cinit, false, false);

#pragma unroll
        for (int v = 0; v < 8; ++v) {
            float d2 = c[v];                 // full distance from the WMMA
            if (d2 < minv[v]) { minv[v] = d2; mini[v] = col; }
        }
    }

    // reduce (min, argmin) across the 16 lanes of each half-wave
#pragma unroll
    for (int s = 1; s < 16; s <<= 1) {
#pragma unroll
        for (int v = 0; v < 8; ++v) {
            float ov = __shfl_xor(minv[v], s, 32);
            int   oi = __shfl_xor(mini[v], s, 32);
            if (ov < minv[v] || (ov == minv[v] && oi < mini[v])) {
                minv[v] = ov; mini[v] = oi;
            }
        }
    }

    if (l == 0) {
        const int base = b * N + i0 + 8 * hi;
#pragma unroll
        for (int v = 0; v < 8; ++v) {
            diff[base + v] = minv[v];
            idxb[base + v] = mini[v];
        }
    }
}

// ---------------------------------------------------------------------------
// Kernel 2: k-th order statistic of diff (threshold m), one block per batch.
// Binary search over the order-preserving uint mapping of float.
// ---------------------------------------------------------------------------
__device__ __forceinline__ unsigned f2u_mono(float f) {
    unsigned u = __float_as_uint(f);
    return (u & 0x80000000u) ? ~u : (u | 0x80000000u);
}
__device__ __forceinline__ float u2f_mono(unsigned u) {
    unsigned bits = (u & 0x80000000u) ? (u & 0x7FFFFFFFu) : ~u;
    return __uint_as_float(bits);
}

__global__ void select_kth_kernel(const float* __restrict__ diff,
                                  float* __restrict__ mth, int N, int kidx) {
    const int b = blockIdx.x;
    const float* d = diff + (size_t)b * N;
    __shared__ unsigned scnt, slo, shi;
    if (threadIdx.x == 0) { slo = 0u; shi = 0xFFFFFFFFu; }
    __syncthreads();
    for (int it = 0; it < 32; ++it) {
        unsigned lo = slo, hi = shi;
        if (lo >= hi) break;                 // uniform across block
        unsigned mid = lo + ((hi - lo) >> 1);
        if (threadIdx.x == 0) scnt = 0u;
        __syncthreads();
        unsigned cnt = 0;
        for (int i = threadIdx.x; i < N; i += blockDim.x)
            cnt += (f2u_mono(d[i]) <= mid) ? 1u : 0u;
        atomicAdd(&scnt, cnt);
        __syncthreads();
        if (threadIdx.x == 0) {
            if (scnt >= (unsigned)(kidx + 1)) shi = mid; else slo = mid + 1u;
        }
        __syncthreads();
    }
    if (threadIdx.x == 0) mth[b] = fmaxf(u2f_mono(slo), 1.0f);
}

// ---------------------------------------------------------------------------
// Kernel 3: per-point Mahalanobis loss + weighted moment accumulation.
// ---------------------------------------------------------------------------
__device__ __forceinline__ void span_cov3(const float cp[7], float C[3][3]) {
    float e0 = cp[0];
    float e1 = e0 + cp[1];
    float e2 = e1 + cp[2];
    float qx = cp[3], qy = cp[4], qz = cp[5], qw = cp[6];
    float n = sqrtf(qx * qx + qy * qy + qz * qz + qw * qw) + 1e-9f;
    float x = qx / n, y = qy / n, z = qz / n, w = qw / n;
    float E[3][3];
    E[0][0] = 1.0f - 2.0f * (y * y + z * z);
    E[0][1] = 2.0f * (x * y - z * w);
    E[0][2] = 2.0f * (x * z + y * w);
    E[1][0] = 2.0f * (x * y + z * w);
    E[1][1] = 1.0f - 2.0f * (x * x + z * z);
    E[1][2] = 2.0f * (y * z - x * w);
    E[2][0] = 2.0f * (x * z - y * w);
    E[2][1] = 2.0f * (y * z + x * w);
    E[2][2] = 1.0f - 2.0f * (x * x + y * y);
    float e[3] = {e0, e1, e2};
#pragma unroll
    for (int i = 0; i < 3; ++i)
#pragma unroll
        for (int k = 0; k < 3; ++k) {
            float s = 0.0f;
#pragma unroll
            for (int j = 0; j < 3; ++j) s += E[i][j] * e[j] * E[k][j];
            C[i][k] = s;
        }
}

#define NACC 19
__global__ void accum_kernel(const float* __restrict__ xyz_pred,
                             const float* __restrict__ xyz_target,
                             const float* __restrict__ cov_pred,
                             const float* __restrict__ cov_target,
                             const float* __restrict__ R_pred,
                             const float* __restrict__ normal_pred,
                             const float* __restrict__ diff,
                             const int* __restrict__ idxb,
                             const float* __restrict__ mth,
                             float* __restrict__ acc,
                             int N, int M) {
    const int b = blockIdx.y;
    const int i = blockIdx.x * blockDim.x + threadIdx.x;
    __shared__ float sacc[NACC];
    if (threadIdx.x < NACC) sacc[threadIdx.x] = 0.0f;
    __syncthreads();

    if (i < N) {
        const float dmin = diff[b * N + i];
        if (dmin < mth[b]) {
            const int j = idxb[b * N + i];
            float xpv[3], xav[3], cp7[7], ct7[7], nrm[3], R[3][3];
#pragma unroll
            for (int k = 0; k < 3; ++k) {
                xpv[k] = xyz_pred[((size_t)b * N + i) * 3 + k];
                xav[k] = xyz_target[((size_t)b * M + j) * 3 + k];
                nrm[k] = normal_pred[((size_t)b * N + i) * 3 + k];
            }
#pragma unroll
            for (int k = 0; k < 7; ++k) {
                cp7[k] = cov_pred[((size_t)b * N + i) * 7 + k];
                ct7[k] = cov_target[((size_t)b * M + j) * 7 + k];
            }
#pragma unroll
            for (int r = 0; r < 3; ++r)
#pragma unroll
                for (int c = 0; c < 3; ++c) R[r][c] = R_pred[b * 9 + r * 3 + c];

            float covp[3][3], cpa[3][3];
            span_cov3(cp7, covp);
            span_cov3(ct7, cpa);

            // sigma = covp + R * cpa * R^T
            float T[3][3], S[3][3];
#pragma unroll
            for (int r = 0; r < 3; ++r)
#pragma unroll
                for (int c = 0; c < 3; ++c) {
                    float s = 0.0f;
#pragma unroll
                    for (int k = 0; k < 3; ++k) s += R[r][k] * cpa[k][c];
                    T[r][c] = s;
                }
#pragma unroll
            for (int r = 0; r < 3; ++r)
#pragma unroll
                for (int c = 0; c < 3; ++c) {
                    float s = 0.0f;
#pragma unroll
                    for (int k = 0; k < 3; ++k) s += T[r][k] * R[c][k];
                    S[r][c] = covp[r][c] + s;
                }

            // adjugate + det, sq = dv^T S^-1 dv
            float dv[3] = {xpv[0] - xav[0], xpv[1] - xav[1], xpv[2] - xav[2]};
            float adj[3][3];
            adj[0][0] =  S[1][1] * S[2][2] - S[1][2] * S[2][1];
            adj[1][0] = -(S[1][0] * S[2][2] - S[1][2] * S[2][0]);
            adj[2][0] =  S[1][0] * S[2][1] - S[1][1] * S[2][0];
            adj[0][1] = -(S[0][1] * S[2][2] - S[0][2] * S[2][1]);
            adj[1][1] =  S[0][0] * S[2][2] - S[0][2] * S[2][0];
            adj[2][1] = -(S[0][0] * S[2][1] - S[0][1] * S[2][0]);
            adj[0][2] =  S[0][1] * S[1][2] - S[0][2] * S[1][1];
            adj[1][2] = -(S[0][0] * S[1][2] - S[0][2] * S[1][0]);
            adj[2][2] =  S[0][0] * S[1][1] - S[0][1] * S[1][0];
            float det = S[0][0] * adj[0][0] + S[0][1] * adj[1][0] + S[0][2] * adj[2][0];
            float quad = 0.0f;
#pragma unroll
            for (int r = 0; r < 3; ++r)
#pragma unroll
                for (int c = 0; c < 3; ++c) quad += dv[r] * adj[r][c] * dv[c];
            const float sq = quad / det;
            const float logdet = logf(det);

            // cosine-similarity weight (diff_na = xa - xp = -dv)
            float dot = -(nrm[0] * dv[0] + nrm[1] * dv[1] + nrm[2] * dv[2]);
            float nn = sqrtf(nrm[0] * nrm[0] + nrm[1] * nrm[1] + nrm[2] * nrm[2]);
            float nd = sqrtf(dv[0] * dv[0] + dv[1] * dv[1] + dv[2] * dv[2]);
            float cs = dot / fmaxf(nn * nd, EPS_COS);
            float w = cs * cs;

            float vals[NACC];
            vals[0] = sq;
            vals[1] = 0.5f * logdet;
            vals[2] = 1.0f;
            vals[3] = w;
#pragma unroll
            for (int k = 0; k < 3; ++k) { vals[4 + k] = w * xpv[k]; vals[7 + k] = w * xav[k]; }
#pragma unroll
            for (int r = 0; r < 3; ++r)
#pragma unroll
                for (int c = 0; c < 3; ++c) vals[10 + r * 3 + c] = w * xpv[r] * xav[c];
#pragma unroll
            for (int k = 0; k < NACC; ++k) atomicAdd(&sacc[k], vals[k]);
        }
    }
    __syncthreads();
    if (threadIdx.x < NACC) atomicAdd(&acc[b * 32 + threadIdx.x], sacc[threadIdx.x]);
}

// ---------------------------------------------------------------------------
// Kernel 4: per-batch 3x3 Kabsch (Jacobi eigen-SVD of H) + loss combine.
// ---------------------------------------------------------------------------
__device__ void jacobi3(float S[3][3], float V[3][3]) {
#pragma unroll
    for (int r = 0; r < 3; ++r)
#pragma unroll
        for (int c = 0; c < 3; ++c) V[r][c] = (r == c) ? 1.0f : 0.0f;
    const int PP[3] = {0, 0, 1};
    const int QQ[3] = {1, 2, 2};
    for (int sweep = 0; sweep < 12; ++sweep) {
        for (int pi = 0; pi < 3; ++pi) {
            int p = PP[pi], q = QQ[pi];
            float apq = S[p][q];
            if (fabsf(apq) > 1e-20f) {
                float theta = (S[q][q] - S[p][p]) / (2.0f * apq);
                float t = copysignf(1.0f, theta) /
                          (fabsf(theta) + sqrtf(1.0f + theta * theta));
                float c = 1.0f / sqrtf(1.0f + t * t);
                float s = t * c;
                for (int k = 0; k < 3; ++k) {
                    float a = S[k][p], bq = S[k][q];
                    S[k][p] = c * a - s * bq;
                    S[k][q] = s * a + c * bq;
                }
                for (int k = 0; k < 3; ++k) {
                    float a = S[p][k], bq = S[q][k];
                    S[p][k] = c * a - s * bq;
                    S[q][k] = s * a + c * bq;
                }
                for (int k = 0; k < 3; ++k) {
                    float a = V[k][p], bq = V[k][q];
                    V[k][p] = c * a - s * bq;
                    V[k][q] = s * a + c * bq;
                }
            }
        }
    }
}

__global__ void finalize_kernel(const float* __restrict__ acc,
                                float* __restrict__ out, int B_) {
    __shared__ float lsum[8];
    const int b = threadIdx.x;
    if (b < B_) {
        const float* a = acc + b * 32;
        const float nvalid = a[2];
        lsum[b] = (a[0] + REG_W * a[1]) / nvalid;

        const float W = a[3];
        float sc[3], tc[3];
#pragma unroll
        for (int k = 0; k < 3; ++k) { sc[k] = a[4 + k] / W; tc[k] = a[7 + k] / W; }
        float H[3][3];
#pragma unroll
        for (int r = 0; r < 3; ++r)
#pragma unroll
            for (int c = 0; c < 3; ++c)
                H[r][c] = a[10 + r * 3 + c] - W * sc[r] * tc[c];

        // eigen-decomposition of H^T H
        float HtH[3][3];
        for (int r = 0; r < 3; ++r)
            for (int c = 0; c < 3; ++c) {
                float s = 0.0f;
                for (int k = 0; k < 3; ++k) s += H[k][r] * H[k][c];
                HtH[r][c] = s;
            }
        float V[3][3];
        jacobi3(HtH, V);
        float lam[3] = {HtH[0][0], HtH[1][1], HtH[2][2]};
        int ord[3] = {0, 1, 2};
        if (lam[ord[0]] < lam[ord[1]]) { int t = ord[0]; ord[0] = ord[1]; ord[1] = t; }
        if (lam[ord[1]] < lam[ord[2]]) { int t = ord[1]; ord[1] = ord[2]; ord[2] = t; }
        if (lam[ord[0]] < lam[ord[1]]) { int t = ord[0]; ord[0] = ord[1]; ord[1] = t; }

        float Vs[3][3], U[3][3], sig[3];
        for (int k = 0; k < 3; ++k) {
            sig[k] = sqrtf(fmaxf(lam[ord[k]], 0.0f));
            for (int r = 0; r < 3; ++r) Vs[r][k] = V[r][ord[k]];
        }
        for (int k = 0; k < 3; ++k) {
            if (sig[k] > 1e-12f) {
                for (int r = 0; r < 3; ++r) {
                    float s = 0.0f;
                    for (int c = 0; c < 3; ++c) s += H[r][c] * Vs[c][k];
                    U[r][k] = s / sig[k];
                }
            } else {
                // degenerate: complete with cross product of previous columns
                float u0[3] = {U[0][0], U[1][0], U[2][0]};
                float u1[3] = {U[0][1], U[1][1], U[2][1]};
                float cx = u0[1] * u1[2] - u0[2] * u1[1];
                float cy = u0[2] * u1[0] - u0[0] * u1[2];
                float cz = u0[0] * u1[1] - u0[1] * u1[0];
                float nn = sqrtf(cx * cx + cy * cy + cz * cz) + 1e-20f;
                U[0][k] = cx / nn; U[1][k] = cy / nn; U[2][k] = cz / nn;
            }
        }

        // d = det(Vs * U^T)
        float Mm[3][3];
        for (int r = 0; r < 3; ++r)
            for (int c = 0; c < 3; ++c) {
                float s = 0.0f;
                for (int k = 0; k < 3; ++k) s += Vs[r][k] * U[c][k];
                Mm[r][c] = s;
            }
        float d = Mm[0][0] * (Mm[1][1] * Mm[2][2] - Mm[1][2] * Mm[2][1])
                - Mm[0][1] * (Mm[1][0] * Mm[2][2] - Mm[1][2] * Mm[2][0])
                + Mm[0][2] * (Mm[1][0] * Mm[2][1] - Mm[1][1] * Mm[2][0]);

        float dk[3] = {1.0f, 1.0f, d};
        float Rres[3][3];
        for (int r = 0; r < 3; ++r)
            for (int c = 0; c < 3; ++c) {
                float s = 0.0f;
                for (int k = 0; k < 3; ++k) s += Vs[r][k] * dk[k] * U[c][k];
                Rres[r][c] = s;
            }
        float tres[3];
        for (int r = 0; r < 3; ++r) {
            float s = 0.0f;
            for (int c = 0; c < 3; ++c) s += Rres[r][c] * sc[c];
            tres[r] = tc[r] - s;
        }

        for (int r = 0; r < 3; ++r)
            for (int c = 0; c < 3; ++c) out[1 + b * 9 + r * 3 + c] = Rres[r][c];
        for (int r = 0; r < 3; ++r) out[1 + B_ * 9 + b * 3 + r] = tres[r];
    }
    __syncthreads();
    if (threadIdx.x == 0) {
        float s = 0.0f;
        for (int k = 0; k < B_; ++k) s += lsum[k];
        out[0] = LOSS_W * s;
    }
}

// ---------------------------------------------------------------------------
extern "C" void kernel_launch(void* const* d_in, const int* in_sizes, int n_in,
                              void* d_out, int out_size, void* d_ws, size_t ws_size,
                              hipStream_t stream) {
    (void)n_in; (void)out_size; (void)ws_size;
    const float* xyz_pred    = (const float*)d_in[0];
    const float* xyz_target  = (const float*)d_in[1];
    const float* cov_pred    = (const float*)d_in[2];
    const float* cov_target  = (const float*)d_in[3];
    const float* R_pred      = (const float*)d_in[4];
    /* d_in[5] = t_pred (unused by reference loss) */
    const float* normal_pred = (const float*)d_in[6];
    /* d_in[7] = normal_target (unused) */

    const int B_ = NBATCH;
    const int N_ = in_sizes[0] / (B_ * 3);
    const int M_ = in_sizes[1] / (B_ * 3);

    char* ws = (char*)d_ws;
    float* diff = (float*)ws;  ws += sizeof(float) * (size_t)B_ * N_;
    int*   idxb = (int*)ws;    ws += sizeof(int) * (size_t)B_ * N_;
    float* mth  = (float*)ws;  ws += sizeof(float) * 8;
    // align accumulator block
    uintptr_t p = (uintptr_t)ws;
    p = (p + 255) & ~(uintptr_t)255;
    float* acc = (float*)p;

    float* out = (float*)d_out;

    init_acc_kernel<<<1, 64, 0, stream>>>(acc, B_ * 32);

    dim3 g1(N_ / (16 * CH_WAVES), B_);
    size_t shbytes = (size_t)(4 * M_) * sizeof(float);   // 3*M xt + M norms
    chamfer_wmma_kernel<<<g1, 256, shbytes, stream>>>(
        xyz_pred, xyz_target, diff, idxb, N_, M_);

    int kidx = (int)((double)N_ * 0.95);
    select_kth_kernel<<<B_, 256, 0, stream>>>(diff, mth, N_, kidx);

    dim3 g3((N_ + 255) / 256, B_);
    accum_kernel<<<g3, 256, 0, stream>>>(
        xyz_pred, xyz_target, cov_pred, cov_target, R_pred, normal_pred,
        diff, idxb, mth, acc, N_, M_);

    finalize_kernel<<<1, 64, 0, stream>>>(acc, out, B_);
}